// VolumetricCELoss_multi_stage_32177894981936
// MI455X (gfx1250) — compile-verified
//
#include <hip/hip_runtime.h>
#include <hip/hip_bf16.h>

// Problem constants (from reference setup_inputs)
#define SS 2
#define BB 8
#define JJ 17
#define XX 64
#define NROW (XX * XX * XX)     // 262144 floats per (s,b,j) row (1 MB)
#define ROWS (SS * BB * JJ)     // 272 rows
#define TILE 4096               // floats per TDM tile (16 KB)
#define NTILES (NROW / TILE)    // 64 tiles per row
#define THREADS 256             // 8 waves (wave32)
#define PT (TILE / THREADS)     // 16 floats per thread per tile
#define FIN_THREADS 288

#define LOG2E 1.4426950408889634f
#define LN2   0.6931471805599453f

typedef unsigned int u32x4 __attribute__((ext_vector_type(4)));
typedef int          i32x8 __attribute__((ext_vector_type(8)));
typedef int          i32x4 __attribute__((ext_vector_type(4)));

// Issue one TDM DMA: copy `nelem` contiguous f32 from global `gptr` into LDS
// at byte offset `lds_byte_addr`. Tracked by TENSORcnt on the issuing wave.
__device__ __forceinline__ void tdm_load_1d(unsigned lds_byte_addr,
                                            const float* gptr, int nelem) {
  unsigned long long ga = (unsigned long long)(uintptr_t)gptr;

  // D# group 0: [1:0]=count=1, [63:32]=lds_addr, [120:64]=global_addr, [127:126]=type=2
  u32x4 g0;
  g0[0] = 1u;
  g0[1] = lds_byte_addr;
  g0[2] = (unsigned)(ga & 0xFFFFFFFFull);
  g0[3] = (unsigned)((ga >> 32) & 0x1FFFFFFull) | (2u << 30);

  // D# group 1: workgroup_mask=0, data_size=2 (4B), no barrier/iterate/pad.
  // tensor_dim0 = nelem, tensor_dim1 = 1, tile_dim0 = nelem, tile_dim1 = 1,
  // tile_dim2 = 0 (unused), tensor_dim0_stride = nelem.
  i32x8 g1;
  g1[0] = 0x00020000;                             // data_size=2 at bits [17:16]
  g1[1] = (nelem & 0xFFFF) << 16;                 // tensor_dim0[15:0] @ bits 63:48
  g1[2] = ((nelem >> 16) & 0xFFFF) | (1 << 16);   // tensor_dim0[31:16] | tensor_dim1=1
  g1[3] = (nelem & 0xFFFF) << 16;                 // tile_dim0 @ bits 127:112
  g1[4] = 1;                                      // tile_dim1=1, tile_dim2=0
  g1[5] = nelem;                                  // tensor_dim0_stride[31:0]
  g1[6] = 0;
  g1[7] = 0;

  i32x4 gz4 = {0, 0, 0, 0};                       // groups 2/3 unused (<=2D)
  i32x8 gz8 = {0, 0, 0, 0, 0, 0, 0, 0};           // 6-arg form: extra group
  __builtin_amdgcn_tensor_load_to_lds(g0, g1, gz4, gz4, gz8, 0);
}

// Merge two (max, sum-of-exp) partials of a logsumexp.
__device__ __forceinline__ void lse_merge(float& M, float& S, float m2, float s2) {
  float nm = fmaxf(M, m2);
  S = S * exp2f((M - nm) * LOG2E) + s2 * exp2f((m2 - nm) * LOG2E);
  M = nm;
}

// One block per (s,b,j) row: streaming single-pass logsumexp over 262144 f32,
// TDM double-buffered through LDS. Writes lse[row] (natural log) to workspace.
__global__ __launch_bounds__(THREADS) void lse_rows_kernel(
    const float* __restrict__ volumes, float* __restrict__ lse_out) {
  __shared__ __align__(16) float tiles[2 * TILE];   // 32 KB double buffer
  __shared__ float redM[THREADS / 32];
  __shared__ float redS[THREADS / 32];

  const int row = blockIdx.x;
  const float* rowp = volumes + (size_t)row * NROW;
  const unsigned ldsbase = (unsigned)(uintptr_t)(&tiles[0]);

  // Scalar (SGPR) condition: only wave 0 issues TDM ops. TDM ignores EXEC, so
  // this must be a scalar branch, which readfirstlane guarantees.
  const bool issuer = (__builtin_amdgcn_readfirstlane((int)(threadIdx.x >> 5)) == 0);

  if (issuer) tdm_load_1d(ldsbase, rowp, TILE);     // prefetch tile 0

  float M = -3.0e38f;
  float S = 0.0f;

  for (int t = 0; t < NTILES; ++t) {
    // Issue tile t+1 into the buffer processed in iteration t-1 (freed by the
    // end-of-loop barrier), then wait for tile t (in-order completion).
    if (t + 1 < NTILES) {
      if (issuer)
        tdm_load_1d(ldsbase + (unsigned)((t + 1) & 1) * (TILE * 4u),
                    rowp + (size_t)(t + 1) * TILE, TILE);
      __builtin_amdgcn_s_wait_tensorcnt(1);
    } else {
      __builtin_amdgcn_s_wait_tensorcnt(0);
    }
    __syncthreads();   // tile t visible to all waves

    const float4* lbuf = (const float4*)(&tiles[(t & 1) * TILE]);
    float v[PT];
#pragma unroll
    for (int k = 0; k < PT / 4; ++k) {             // ds_load_b128 x4
      float4 q = lbuf[k * THREADS + threadIdx.x];
      v[4 * k + 0] = q.x; v[4 * k + 1] = q.y;
      v[4 * k + 2] = q.z; v[4 * k + 3] = q.w;
    }
    float m = v[0];
#pragma unroll
    for (int i = 1; i < PT; ++i) m = fmaxf(m, v[i]);
    float se = 0.0f;
#pragma unroll
    for (int i = 0; i < PT; ++i) se += exp2f((v[i] - m) * LOG2E);  // 1 exp2/elem
    lse_merge(M, S, m, se);

    __syncthreads();   // everyone done reading buf[t&1] before it is refilled
  }

  // wave32 butterfly reduction of (M, S)
  for (int off = 16; off > 0; off >>= 1) {
    float m2 = __shfl_xor(M, off, 32);
    float s2 = __shfl_xor(S, off, 32);
    lse_merge(M, S, m2, s2);
  }
  if ((threadIdx.x & 31) == 0) {
    redM[threadIdx.x >> 5] = M;
    redS[threadIdx.x >> 5] = S;
  }
  __syncthreads();
  if (threadIdx.x == 0) {
    float Mt = redM[0], St = redS[0];
    for (int w = 1; w < THREADS / 32; ++w) lse_merge(Mt, St, redM[w], redS[w]);
    lse_out[row] = Mt + LN2 * log2f(St);           // natural-log LSE
  }
}

// Single block: quantize labels, gather one logit per row, per-stage fixed-order
// reduction (deterministic), in-bounds gate, scalar output.
__global__ __launch_bounds__(FIN_THREADS) void finalize_kernel(
    const float* __restrict__ volumes, const float* __restrict__ label,
    const float* __restrict__ vmax_cat, const float* __restrict__ vmin_cat,
    const float* __restrict__ lse, float* __restrict__ out) {
  __shared__ float c_sh[ROWS];
  __shared__ int mx_sh[ROWS];
  __shared__ int mn_sh[ROWS];

  const int r = threadIdx.x;
  if (r < ROWS) {
    const int s = r / (BB * JJ);
    const int rem = r % (BB * JJ);
    const int b = rem / JJ;
    const int j = rem % JJ;
    int idx[3];
#pragma unroll
    for (int c = 0; c < 3; ++c) {
      float vmx = vmax_cat[(b * SS + s) * 3 + c];
      float vmn = vmin_cat[(b * SS + s) * 3 + c];
      float mean = 0.5f * (vmx + vmn);
      float scl  = 0.5f * (vmx - vmn);
      float g = (label[(b * JJ + j) * 3 + c] - mean) / scl;
      idx[c] = (int)floorf((g + 1.0f) * 0.5f * (float)(XX - 1));
    }
    int flat = idx[0] * (XX * XX) + idx[1] * XX + idx[2];
    float logit = volumes[(size_t)r * NROW + flat];
    float p = exp2f((logit - lse[r]) * LOG2E);
    c_sh[r]  = 0.01f * (-logf(p + 1e-6f));         // BETA * -log(p + EPS)
    mx_sh[r] = max(idx[0], max(idx[1], idx[2]));
    mn_sh[r] = min(idx[0], min(idx[1], idx[2]));
  }
  __syncthreads();

  if (threadIdx.x == 0) {
    float total = 0.0f;
    for (int s = 0; s < SS; ++s) {
      float acc = 0.0f;
      int mx = -2147483647, mn = 2147483647;
      for (int k = 0; k < BB * JJ; ++k) {          // fixed order: deterministic
        int rr = s * (BB * JJ) + k;
        acc += c_sh[rr];
        mx = max(mx, mx_sh[rr]);
        mn = min(mn, mn_sh[rr]);
      }
      bool inb = (mx < XX) && (mx > 0) && (mn < XX) && (mn > 0);
      total += inb ? (acc / (float)(BB * JJ)) : 0.0f;
    }
    out[0] = total;
  }
}

extern "C" void kernel_launch(void* const* d_in, const int* in_sizes, int n_in,
                              void* d_out, int out_size, void* d_ws, size_t ws_size,
                              hipStream_t stream) {
  const float* volumes  = (const float*)d_in[0];
  const float* label    = (const float*)d_in[1];
  const float* vmax_cat = (const float*)d_in[2];
  const float* vmin_cat = (const float*)d_in[3];
  float* lse = (float*)d_ws;                       // ROWS floats of scratch
  float* out = (float*)d_out;

  lse_rows_kernel<<<ROWS, THREADS, 0, stream>>>(volumes, lse);
  finalize_kernel<<<1, FIN_THREADS, 0, stream>>>(volumes, label, vmax_cat,
                                                 vmin_cat, lse, out);
}